// WFMLayer_25555055411604
// MI455X (gfx1250) — compile-verified
//
#include <hip/hip_runtime.h>

// ---------------------------------------------------------------------------
// Weighted Fréchet mean of K=2 SPD 32x32 matrices, 10 fixed-point iterations.
// GEMM-only matrix functions (Newton-Schulz sqrt/invsqrt, inverse scaling-
// squaring log, scaling-squaring exp) -> v_wmma_f32_16x16x4_f32 everywhere.
// One wave32 owns one output matrix. LDS addressed by integer offsets into the
// extern __shared__ base (stays addrspace(3) -> ds ops). GEMMs are phased:
// B operands stay register-resident, A/acc processed per 16-row half to keep
// peak VGPR pressure ~64 regs (no spills under the 256-reg cap).
// 8 matrices/wave (144 KB / 4-wave WG) -> 2 workgroups per 320 KB WGP.
// ---------------------------------------------------------------------------

typedef __attribute__((ext_vector_type(2))) float v2f;
typedef __attribute__((ext_vector_type(8))) float v8f;

typedef __attribute__((address_space(1))) const float gcf;  // global const
typedef __attribute__((address_space(1))) float gf;         // global

#define LDST 36                 // padded row stride (floats)
#define MATSZ (32 * LDST)       // 1152 floats per 32x32 matrix
#define NMAT 8                  // scratch matrices per wave
#define WAVES_PER_BLOCK 4
#define N_OUT (8 * 127 * 9)     // 9144 output matrices
#define NS_MEAN_ITERS 12        // even
#define NS_LOG_ITERS 10         // even
#define S_LOG 5                 // inverse scaling-squaring depth (2^5)
#define J_LOG 8                 // Mercator terms
#define J_EXP 8                 // Taylor terms
#define S_EXP 5                 // exp squarings (2^5)

#define WMMA_F32(a, b, c) \
  __builtin_amdgcn_wmma_f32_16x16x4_f32(false, (a), false, (b), (short)0, (c), \
                                        false, false)

// Store one 16x16 f32 C/D tile (v8f layout per ISA 7.12.2).
template <bool ACCUM>
__device__ __forceinline__ void store_tile(float* sm, unsigned C, const v8f& acc,
                                           int ti, int tj, float alpha,
                                           float diag_add, int half, int l15) {
  #pragma unroll
  for (int r = 0; r < 8; ++r) {
    const int row = ti + r + 8 * half;
    const int col = tj + l15;
    float v = alpha * acc[r];
    v += (row == col) ? diag_add : 0.0f;
    if (ACCUM) v += sm[C + (unsigned)(row * LDST + col)];
    sm[C + (unsigned)(row * LDST + col)] = v;
  }
}

// C = alpha * (A @ B) + diag_add * I  (+= C when ACCUM). Offsets into smem.
template <bool ACCUM>
__device__ __forceinline__ void wave_gemm(float* sm, unsigned A, unsigned B,
                                          unsigned C, float alpha,
                                          float diag_add, unsigned lane) {
  const int half = (int)(lane >> 4);   // 0: K pair {0,1}, 1: K pair {2,3}
  const int l15 = (int)(lane & 15u);
  v2f b0[8], b1[8];                    // B resident for both row-halves
  #pragma unroll
  for (int k = 0; k < 8; ++k) {
    const int kk = 4 * k + 2 * half;
    b0[k].x = sm[B + (unsigned)(kk * LDST + l15)];
    b0[k].y = sm[B + (unsigned)((kk + 1) * LDST + l15)];
    b1[k].x = sm[B + (unsigned)(kk * LDST + 16 + l15)];
    b1[k].y = sm[B + (unsigned)((kk + 1) * LDST + 16 + l15)];
  }
  #pragma unroll
  for (int ti = 0; ti < 32; ti += 16) {
    v2f a[8];
    #pragma unroll
    for (int k = 0; k < 8; ++k) {
      const int kk = 4 * k + 2 * half;
      a[k].x = sm[A + (unsigned)((ti + l15) * LDST + kk)];
      a[k].y = sm[A + (unsigned)((ti + l15) * LDST + kk + 1)];
    }
    v8f c0 = {}, c1 = {};
    #pragma unroll
    for (int k = 0; k < 8; ++k) {      // 2 independent chains per k-step
      c0 = WMMA_F32(a[k], b0[k], c0);
      c1 = WMMA_F32(a[k], b1[k], c1);
    }
    store_tile<ACCUM>(sm, C, c0, ti, 0, alpha, diag_add, half, l15);
    store_tile<ACCUM>(sm, C, c1, ti, 16, alpha, diag_add, half, l15);
  }
}

// C = alpha * (A @ Bg), B streamed from global (row-major, stride 32, L2-hot).
__device__ __forceinline__ void wave_gemm_gB(float* sm, unsigned A,
                                             gcf* __restrict__ Bg, unsigned C,
                                             float alpha, unsigned lane) {
  const int half = (int)(lane >> 4);
  const int l15 = (int)(lane & 15u);
  v2f b0[8], b1[8];
  #pragma unroll
  for (int k = 0; k < 8; ++k) {
    const int kk = 4 * k + 2 * half;
    b0[k].x = Bg[kk * 32 + l15];
    b0[k].y = Bg[(kk + 1) * 32 + l15];
    b1[k].x = Bg[kk * 32 + 16 + l15];
    b1[k].y = Bg[(kk + 1) * 32 + 16 + l15];
  }
  #pragma unroll
  for (int ti = 0; ti < 32; ti += 16) {
    v2f a[8];
    #pragma unroll
    for (int k = 0; k < 8; ++k) {
      const int kk = 4 * k + 2 * half;
      a[k].x = sm[A + (unsigned)((ti + l15) * LDST + kk)];
      a[k].y = sm[A + (unsigned)((ti + l15) * LDST + kk + 1)];
    }
    v8f c0 = {}, c1 = {};
    #pragma unroll
    for (int k = 0; k < 8; ++k) {
      c0 = WMMA_F32(a[k], b0[k], c0);
      c1 = WMMA_F32(a[k], b1[k], c1);
    }
    store_tile<false>(sm, C, c0, ti, 0, alpha, 0.0f, half, l15);
    store_tile<false>(sm, C, c1, ti, 16, alpha, 0.0f, half, l15);
  }
}

__device__ __forceinline__ void wave_copy_scale(float* sm, unsigned dst,
                                                unsigned src, float s,
                                                unsigned lane) {
  for (int i = (int)lane; i < 1024; i += 32) {
    const unsigned idx = (unsigned)((i >> 5) * LDST + (i & 31));
    sm[dst + idx] = s * sm[src + idx];
  }
}

__device__ __forceinline__ void wave_set_identity(float* sm, unsigned dst,
                                                  float d, unsigned lane) {
  for (int i = (int)lane; i < 1024; i += 32) {
    const int r = i >> 5, c = i & 31;
    sm[dst + (unsigned)(r * LDST + c)] = (r == c) ? d : 0.0f;
  }
}

__device__ __forceinline__ float wave_fro(const float* sm, unsigned A,
                                          unsigned lane) {
  float s = 0.0f;
  for (int i = (int)lane; i < 1024; i += 32) {
    const float v = sm[A + (unsigned)((i >> 5) * LDST + (i & 31))];
    s += v * v;
  }
  #pragma unroll
  for (int off = 16; off > 0; off >>= 1) s += __shfl_xor(s, off, 32);
  return sqrtf(s);
}

// Coupled Newton-Schulz: Y ~= sqrt(A/f), Z ~= invsqrt(A/f), f = ||A||_F.
// iters must be even so results land at offsets Y, Z. A is consumed before
// any scratch write, so callers may alias W1 (first yn) with A.
__device__ __forceinline__ float ns_sqrt_invsqrt(float* sm, unsigned A,
                                                 unsigned Y, unsigned Z,
                                                 unsigned W1, unsigned W2,
                                                 unsigned W3, int iters,
                                                 unsigned lane) {
  const float f = wave_fro(sm, A, lane) + 1e-30f;
  wave_copy_scale(sm, Y, A, 1.0f / f, lane);
  wave_set_identity(sm, Z, 1.0f, lane);
  unsigned y = Y, yn = W1, z = Z, zn = W2, t = W3;
  #pragma unroll 1
  for (int it = 0; it < iters; ++it) {
    wave_gemm<false>(sm, z, y, t, -0.5f, 1.5f, lane);  // t = 1.5I - 0.5*Z@Y
    wave_gemm<false>(sm, y, t, yn, 1.0f, 0.0f, lane);  // Y' = Y@t
    wave_gemm<false>(sm, t, z, zn, 1.0f, 0.0f, lane);  // Z' = t@Z
    unsigned s;
    s = y; y = yn; yn = s;
    s = z; z = zn; zn = s;
  }
  return f;
}

// G (+)= wk * log(X): inverse scaling-squaring + Mercator-Horner. X destroyed.
// Scratch W1..W4; X itself doubles as NS ping-pong scratch.
template <bool ACCUM>
__device__ __forceinline__ void wave_log_accum(float* sm, unsigned X, float wk,
                                               unsigned G, unsigned W1,
                                               unsigned W2, unsigned W3,
                                               unsigned W4, unsigned lane) {
  #pragma unroll 1
  for (int s = 0; s < S_LOG; ++s) {
    const float f =
        ns_sqrt_invsqrt(sm, X, W1, W2, /*yn=*/X, W3, W4, NS_LOG_ITERS, lane);
    wave_copy_scale(sm, X, W1, sqrtf(f), lane);        // X <- sqrt(X)
  }
  sm[X + lane * LDST + lane] -= 1.0f;                  // X <- E = X - I
  // q(E) = c1 I + E(c2 I + ... + E*cJ I),  cj = (-1)^(j+1)/j
  unsigned acc = W1, accn = W2;
  wave_set_identity(sm, acc, ((J_LOG & 1) ? 1.0f : -1.0f) / (float)J_LOG, lane);
  #pragma unroll 1
  for (int j = J_LOG - 1; j >= 1; --j) {
    const float cj = ((j & 1) ? 1.0f : -1.0f) / (float)j;
    wave_gemm<false>(sm, X, acc, accn, 1.0f, cj, lane);
    const unsigned s = acc; acc = accn; accn = s;
  }
  // log(X_orig) ~= 2^S_LOG * E @ q(E); fold 2^S and Frechet weight into alpha.
  wave_gemm<ACCUM>(sm, X, acc, G, wk * (float)(1 << S_LOG), 0.0f, lane);
}

// exp(Tm) by scaling-squaring; returns offset of result (W1 or W2).
__device__ __forceinline__ unsigned wave_exp(float* sm, unsigned Tm,
                                             unsigned W1, unsigned W2,
                                             unsigned Wa, unsigned lane) {
  wave_copy_scale(sm, Wa, Tm, 1.0f / (float)(1 << S_EXP), lane);
  unsigned acc = W1, accn = W2;
  wave_set_identity(sm, acc, 1.0f / 40320.0f, lane);   // 1/8!
  float cf = 1.0f / 5040.0f;                           // invfact[J_EXP-1]
  #pragma unroll 1
  for (int j = J_EXP - 1; j >= 0; --j) {
    wave_gemm<false>(sm, Wa, acc, accn, 1.0f, cf, lane);
    cf *= (float)j;                                    // invfact[j-1]
    const unsigned s = acc; acc = accn; accn = s;
  }
  #pragma unroll 1
  for (int s = 0; s < S_EXP; ++s) {
    wave_gemm<false>(sm, acc, acc, accn, 1.0f, 0.0f, lane);
    const unsigned t = acc; acc = accn; accn = t;
  }
  return acc;
}

__global__ __launch_bounds__(WAVES_PER_BLOCK * 32, 4)
void wfm_kernel(const float* __restrict__ x, const float* __restrict__ raw_w,
                float* __restrict__ out) {
  extern __shared__ float smem[];
  const unsigned lane = threadIdx.x & 31u;
  const unsigned wave = threadIdx.x >> 5;
  const int gidx = (int)blockIdx.x * WAVES_PER_BLOCK + (int)wave;
  if (gidx >= N_OUT) return;                           // wave-uniform

  // softmax over the two raw weights
  const float r0 = raw_w[0], r1 = raw_w[1];
  const float mx = fmaxf(r0, r1);
  const float e0 = __expf(r0 - mx), e1 = __expf(r1 - mx);
  const float w0 = e0 / (e0 + e1), w1 = e1 / (e0 + e1);

  const int c = gidx % 9;
  const int t = (gidx / 9) % 127;
  const int b = gidx / (9 * 127);
  gcf* M0 = (gcf*)(x + ((((size_t)b * 128 + t) * 9 + c) << 10));
  gcf* M1 = M0 + ((size_t)9 << 10);                    // t+1 window

  const unsigned base = wave * (NMAT * MATSZ);
  const unsigned P0 = base + 0 * MATSZ;                // mean / scratch
  const unsigned P1 = base + 1 * MATSZ;                // Y_inf = sqrt(mean/f)
  const unsigned P2 = base + 2 * MATSZ;                // Z_inf = invsqrt(mean/f)
  const unsigned P3 = base + 3 * MATSZ;
  const unsigned P4 = base + 4 * MATSZ;
  const unsigned P5 = base + 5 * MATSZ;
  const unsigned P6 = base + 6 * MATSZ;                // X (inner_k) / scratch
  const unsigned P7 = base + 7 * MATSZ;                // tangent accumulator G

  // mean = w0*M0 + w1*M1
  for (int i = (int)lane; i < 1024; i += 32)
    smem[P0 + (unsigned)((i >> 5) * LDST + (i & 31))] = w0 * M0[i] + w1 * M1[i];

  #pragma unroll 1
  for (int it = 0; it < 10; ++it) {
    const float f =
        ns_sqrt_invsqrt(smem, P0, P1, P2, P3, P4, P5, NS_MEAN_ITERS, lane);
    const float invf = 1.0f / f;                       // invsqrt(mean)^2 scale

    // inner_0 = invsqrt(mean) @ M0 @ invsqrt(mean) = (Z@M0@Z)/f
    wave_gemm_gB(smem, P2, M0, P3, 1.0f, lane);
    wave_gemm<false>(smem, P3, P2, P6, invf, 0.0f, lane);
    wave_log_accum<false>(smem, P6, w0, P7, P3, P4, P5, P0, lane);

    // inner_1
    wave_gemm_gB(smem, P2, M1, P3, 1.0f, lane);
    wave_gemm<false>(smem, P3, P2, P6, invf, 0.0f, lane);
    wave_log_accum<true>(smem, P6, w1, P7, P3, P4, P5, P0, lane);

    // mean <- sqrt(mean) @ exp(G) @ sqrt(mean) = f * (Y @ exp(G) @ Y)
    const unsigned R = wave_exp(smem, P7, P3, P4, P6, lane);
    wave_gemm<false>(smem, P1, R, P5, 1.0f, 0.0f, lane);
    wave_gemm<false>(smem, P5, P1, P0, f, 0.0f, lane);
  }

  gf* og = (gf*)(out + ((size_t)gidx << 10));
  for (int i = (int)lane; i < 1024; i += 32)
    og[i] = smem[P0 + (unsigned)((i >> 5) * LDST + (i & 31))];
}

extern "C" void kernel_launch(void* const* d_in, const int* in_sizes, int n_in,
                              void* d_out, int out_size, void* d_ws, size_t ws_size,
                              hipStream_t stream) {
  (void)in_sizes; (void)n_in; (void)out_size; (void)d_ws; (void)ws_size;
  const float* x = (const float*)d_in[0];
  const float* raw_w = (const float*)d_in[1];
  float* out = (float*)d_out;
  const int blocks = (N_OUT + WAVES_PER_BLOCK - 1) / WAVES_PER_BLOCK;  // 2286
  const size_t shmem = (size_t)WAVES_PER_BLOCK * NMAT * MATSZ * sizeof(float);
  hipLaunchKernelGGL(wfm_kernel, dim3(blocks), dim3(WAVES_PER_BLOCK * 32),
                     shmem, stream, x, raw_w, out);
}